// MultGroupConv_75703093559753
// MI455X (gfx1250) — compile-verified
//
#include <hip/hip_runtime.h>
#include <hip/hip_bf16.h>

typedef float v2f __attribute__((ext_vector_type(2)));
typedef float v8f __attribute__((ext_vector_type(8)));

#define CIN    128
#define COUT_  128
#define STRIPS 5      // 16-row strips per GEMM block (50000 = 625 * 5 * 16)

// ---------------------------------------------------------------------------
// Kernel A: sum_s[n] = sum_g softmax(x[n] @ W_group)_g   (one wave per node)
// ---------------------------------------------------------------------------
__global__ __launch_bounds__(256)
void group_sum_kernel(const float* __restrict__ x,
                      const float* __restrict__ Wg,
                      float* __restrict__ sum_s, int n_nodes) {
    int lane = threadIdx.x & 31;
    int node = blockIdx.x * (blockDim.x >> 5) + (threadIdx.x >> 5);
    if (node >= n_nodes) return;
    const float* xr = x + (size_t)node * CIN;
    float z0 = 0.f, z1 = 0.f, z2 = 0.f, z3 = 0.f;
#pragma unroll
    for (int i = 0; i < 4; ++i) {
        int c = lane + 32 * i;
        float xv = xr[c];
        const float* w = Wg + c * 4;
        z0 += xv * w[0]; z1 += xv * w[1]; z2 += xv * w[2]; z3 += xv * w[3];
    }
#pragma unroll
    for (int off = 16; off; off >>= 1) {          // wave32 xor-reduce
        z0 += __shfl_xor(z0, off, 32);
        z1 += __shfl_xor(z1, off, 32);
        z2 += __shfl_xor(z2, off, 32);
        z3 += __shfl_xor(z3, off, 32);
    }
    if (lane == 0) {
        float m  = fmaxf(fmaxf(z0, z1), fmaxf(z2, z3));
        float e0 = expf(z0 - m), e1 = expf(z1 - m);
        float e2 = expf(z2 - m), e3 = expf(z3 - m);
        float es = e0 + e1 + e2 + e3;
        sum_s[node] = e0 / es + e1 / es + e2 / es + e3 / es;
    }
}

// ---------------------------------------------------------------------------
// Kernel B: fused WMMA GEMM  H = x @ [W_self | W_nbr]  (fp32, 16x16x4)
//   out[n] = sum_s[n]*(x@W_self)[n] + 4*b    (cols   0..127)
//   msg[n] = sum_s[n]*(x@W_nbr )[n]          (cols 128..255)
// 512 threads = 16 waves; wave owns a 16-col B tile held in 64 VGPRs for the
// whole block; block sweeps STRIPS row-strips with double-buffered LDS A.
// ---------------------------------------------------------------------------
__global__ __launch_bounds__(512)
void gemm_kernel(const float* __restrict__ x,
                 const float* __restrict__ Wself,
                 const float* __restrict__ Wnbr,
                 const float* __restrict__ bias,
                 const float* __restrict__ sum_s,
                 float* __restrict__ out,
                 float* __restrict__ msg) {
    __shared__ float As[2][16 * CIN];   // double-buffered 16x128 A strip
    __shared__ float Ss[2][16];

    const int tid  = threadIdx.x;
    const int wave = tid >> 5;
    const int lane = tid & 31;
    const int cl   = lane & 15;          // col in 16-wide tile / row of A
    const int koff = (lane >> 4) << 1;   // 0 (lanes 0-15) or 2 (lanes 16-31)
    const int n0   = wave * 16;          // combined-column base 0..240

    // B base: first 8 waves -> W_self cols, last 8 waves -> W_nbr cols
    const float* Wb = (n0 < COUT_) ? (Wself + n0) : (Wnbr + (n0 - COUT_));

    // preload the 128x16 B tile once per block (64 f32 per lane)
    float breg[64];
#pragma unroll
    for (int s = 0; s < 32; ++s) {
        int k = 4 * s + koff;
        breg[2 * s]     = Wb[(size_t)k       * COUT_ + cl];
        breg[2 * s + 1] = Wb[(size_t)(k + 1) * COUT_ + cl];
    }
    const float bv = (n0 < COUT_) ? 4.0f * bias[n0 + cl] : 0.0f;  // G*b
    const int mbase = (lane >> 4) << 3;

    for (int it = 0; it < STRIPS; ++it) {
        const int r0  = (blockIdx.x * STRIPS + it) * 16;
        const int buf = it & 1;

        // cooperative A-strip load: 2048 floats = 512 threads x float4
        ((float4*)As[buf])[tid] = ((const float4*)(x + (size_t)r0 * CIN))[tid];
        if (tid < 16) Ss[buf][tid] = sum_s[r0 + tid];
        __syncthreads();

        v8f acc = {};
#pragma unroll
        for (int s = 0; s < 32; ++s) {
            const float* ap = &As[buf][cl * CIN + 4 * s + koff];
            v2f a; a.x = ap[0];       a.y = ap[1];          // ds_load_b64
            v2f b; b.x = breg[2 * s]; b.y = breg[2 * s + 1];
            acc = __builtin_amdgcn_wmma_f32_16x16x4_f32(
                      false, a, false, b, (short)0, acc, false, false);
        }

        // C/D layout: VGPR j holds M=j (lanes 0-15) / M=j+8 (lanes 16-31)
        if (n0 < COUT_) {
#pragma unroll
            for (int j = 0; j < 8; ++j) {
                int m = mbase + j;
                out[(size_t)(r0 + m) * COUT_ + n0 + cl] =
                    acc[j] * Ss[buf][m] + bv;
            }
        } else {
#pragma unroll
            for (int j = 0; j < 8; ++j) {
                int m = mbase + j;
                msg[(size_t)(r0 + m) * COUT_ + (n0 - COUT_) + cl] =
                    acc[j] * Ss[buf][m];
            }
        }
    }
}

// ---------------------------------------------------------------------------
// Kernel C: edge scatter  out[dst] += msg[src]
// one wave per edge, 4 channels per lane: b128 gather + 4 f32 atomics/lane.
// src/dst loads are wave-uniform (scalarized); all traffic is L2-resident.
// ---------------------------------------------------------------------------
__global__ __launch_bounds__(256)
void scatter_kernel(const float* __restrict__ msg,
                    const int* __restrict__ src,
                    const int* __restrict__ dst,
                    float* __restrict__ out, int n_edges) {
    unsigned t = blockIdx.x * 256u + threadIdx.x;
    unsigned e = t >> 5;
    unsigned q = (t & 31u) * 4u;
    if (e >= (unsigned)n_edges) return;
    int s = src[e];
    int d = dst[e];
    const float4 v = *(const float4*)(msg + (size_t)s * COUT_ + q);
    float* o = out + (size_t)d * COUT_ + q;
    atomicAdd(o + 0, v.x);
    atomicAdd(o + 1, v.y);
    atomicAdd(o + 2, v.z);
    atomicAdd(o + 3, v.w);
}

// ---------------------------------------------------------------------------
extern "C" void kernel_launch(void* const* d_in, const int* in_sizes, int n_in,
                              void* d_out, int out_size, void* d_ws, size_t ws_size,
                              hipStream_t stream) {
    const float* x     = (const float*)d_in[0];
    const float* Wg    = (const float*)d_in[1];
    const float* Wself = (const float*)d_in[2];
    const float* Wnbr  = (const float*)d_in[3];
    const float* bias  = (const float*)d_in[4];
    const int*   edge  = (const int*)d_in[5];   // [2, E]
    // d_in[6] (batch), d_in[7] (group_ptr): folded away by the group-sum
    // reduction (softmax rows summed over G before the scatter).

    const int N = in_sizes[0] / CIN;   // 50000
    const int E = in_sizes[5] / 2;     // 250000

    float* out   = (float*)d_out;
    float* msg   = (float*)d_ws;                     // N*COUT f32
    float* sum_s = msg + (size_t)N * COUT_;          // N f32

    // 1) per-node group softmax sums (8 nodes per 256-thread block)
    group_sum_kernel<<<(N + 7) / 8, 256, 0, stream>>>(x, Wg, sum_s, N);

    // 2) fused WMMA GEMM: 625 blocks x 5 strips x 16 rows = 50000 rows
    gemm_kernel<<<N / (16 * STRIPS), 512, 0, stream>>>(
        x, Wself, Wnbr, bias, sum_s, out, msg);

    // 3) neighbor aggregation scatter: one wave32 per edge
    unsigned total = (unsigned)E * 32u;
    scatter_kernel<<<(total + 255u) / 256u, 256, 0, stream>>>(
        msg, edge, edge + E, out, E);
}